// HistogramObserver_46892452937842
// MI455X (gfx1250) — compile-verified
//
#include <hip/hip_runtime.h>

// HistogramObserver for MI455X (gfx1250, wave32).
// init ws -> global min/max -> async-staged LDS-privatized 2048-bin histogram -> combine.
// Memory-bound: 2 passes over 256MiB => ~22us floor @ 23.3 TB/s.
// CDNA5 paths: global_load_async_to_lds_b128 + s_wait_asynccnt (ASYNCcnt),
// ds_add_u32/ds_add_f32 atomics, global_load_b128, global_prefetch_b8,
// wave32 shuffles, ordered-key global_atomic_min/max_u32.

#define BINS 2048
#define TPB  256            // 8 wave32 waves per block
#define WAVES (TPB / 32)
#define MAXBLOCKS 2048

// ---- ordered-key encoding: float -> uint32 that compares like the float ----
__device__ __forceinline__ unsigned f2key(float f) {
  unsigned u = __float_as_uint(f);
  return (u & 0x80000000u) ? ~u : (u | 0x80000000u);
}
__device__ __forceinline__ float key2f(unsigned k) {
  unsigned u = (k & 0x80000000u) ? (k ^ 0x80000000u) : ~k;
  return __uint_as_float(u);
}

// ws layout: [0..BINS-1] u32 histogram counts, [BINS] min key, [BINS+1] max key
__global__ void hobs_init_kernel(unsigned* __restrict__ ws) {
  for (int i = threadIdx.x; i < BINS; i += TPB) ws[i] = 0u;
  if (threadIdx.x == 0) { ws[BINS] = 0xFFFFFFFFu; ws[BINS + 1] = 0u; }
}

__global__ void hobs_minmax_kernel(const float4* __restrict__ x4, int n4,
                                   const float* __restrict__ x, int n,
                                   unsigned* __restrict__ ws) {
  float lmin = __uint_as_float(0x7F800000u);   // +inf
  float lmax = __uint_as_float(0xFF800000u);   // -inf
  const int stride = gridDim.x * blockDim.x;
  const int gid = blockIdx.x * blockDim.x + threadIdx.x;
  for (int i = gid; i < n4; i += stride) {
    if (i + stride < n4) __builtin_prefetch(&x4[i + stride], 0, 0);  // global_prefetch_b8
    float4 v = x4[i];
    lmin = fminf(lmin, fminf(fminf(v.x, v.y), fminf(v.z, v.w)));
    lmax = fmaxf(lmax, fmaxf(fmaxf(v.x, v.y), fmaxf(v.z, v.w)));
  }
  for (int i = (n4 << 2) + gid; i < n; i += stride) {  // tail (n % 4)
    float v = x[i];
    lmin = fminf(lmin, v);
    lmax = fmaxf(lmax, v);
  }
  // wave32 butterfly reduction
  #pragma unroll
  for (int off = 16; off > 0; off >>= 1) {
    lmin = fminf(lmin, __shfl_xor(lmin, off, 32));
    lmax = fmaxf(lmax, __shfl_xor(lmax, off, 32));
  }
  __shared__ float smin[WAVES], smax[WAVES];
  const int wave = threadIdx.x >> 5;
  if ((threadIdx.x & 31) == 0) { smin[wave] = lmin; smax[wave] = lmax; }
  __syncthreads();
  if (threadIdx.x == 0) {
    float bmin = smin[0], bmax = smax[0];
    #pragma unroll
    for (int w = 1; w < WAVES; ++w) {
      bmin = fminf(bmin, smin[w]);
      bmax = fmaxf(bmax, smax[w]);
    }
    atomicMin(&ws[BINS], f2key(bmin));       // global_atomic_min_u32
    atomicMax(&ws[BINS + 1], f2key(bmax));   // global_atomic_max_u32
  }
}

__device__ __forceinline__ int hobs_bin(float v, float mn, float rng, bool ok) {
  if (!ok) return 0;
  // match reference exactly: pos = (v-mn)/rng ; idx = trunc(pos*BINS) clipped
  float pos = (v - mn) / rng;
  int idx = (int)(pos * (float)BINS);
  idx = idx < 0 ? 0 : idx;
  idx = idx > (BINS - 1) ? (BINS - 1) : idx;
  return idx;
}

__device__ __forceinline__ void hobs_async_fetch(unsigned lds_addr,
                                                 const float4* gaddr) {
  // CDNA5 async copy: HBM -> LDS, tracked by ASYNCcnt (CDNA5 ISA)
  asm volatile("global_load_async_to_lds_b128 %0, %1, off"
               :: "v"(lds_addr), "v"(gaddr)
               : "memory");
}

__global__ void hobs_hist_kernel(const float4* __restrict__ x4, int n4,
                                 const float* __restrict__ x, int n,
                                 unsigned* __restrict__ ws) {
  __shared__ unsigned h[BINS];                  // 8KB histogram
  __shared__ float4 stage[2][WAVES][32];        // 8KB double-buffered stage
  for (int i = threadIdx.x; i < BINS; i += TPB) h[i] = 0u;
  const float mn = key2f(ws[BINS]);
  const float mx = key2f(ws[BINS + 1]);
  const float rng = mx - mn;
  const bool ok = rng > 0.0f;
  __syncthreads();

  const int lane = threadIdx.x & 31;
  const int wave = threadIdx.x >> 5;
  const int gwave = blockIdx.x * WAVES + wave;      // global wave id
  const int nwaves = gridDim.x * WAVES;
  const int nchunks = n4 >> 5;                      // chunks of 32 float4 (512B/wave)

  // flat LDS addresses: low 32 bits of the generic pointer are the DS offset
  unsigned lbuf[2];
  lbuf[0] = (unsigned)(size_t)&stage[0][wave][lane];
  lbuf[1] = (unsigned)(size_t)&stage[1][wave][lane];

  int c = gwave;
  if (c < nchunks) hobs_async_fetch(lbuf[0], x4 + ((size_t)c << 5) + lane);
  int buf = 0;
  for (; c < nchunks; c += nwaves) {
    const int cn = c + nwaves;
    if (cn < nchunks) {
      hobs_async_fetch(lbuf[buf ^ 1], x4 + ((size_t)cn << 5) + lane);
      asm volatile("s_wait_asynccnt 0x1" ::: "memory");  // in-order: current buf ready
    } else {
      asm volatile("s_wait_asynccnt 0x0" ::: "memory");
    }
    float4 v = stage[buf][wave][lane];                   // ds_load_b128
    atomicAdd(&h[hobs_bin(v.x, mn, rng, ok)], 1u);       // ds_add_u32
    atomicAdd(&h[hobs_bin(v.y, mn, rng, ok)], 1u);
    atomicAdd(&h[hobs_bin(v.z, mn, rng, ok)], 1u);
    atomicAdd(&h[hobs_bin(v.w, mn, rng, ok)], 1u);
    buf ^= 1;
  }

  // float4s not covered by full chunks, then scalar tail
  const int stride = gridDim.x * blockDim.x;
  const int gid = blockIdx.x * blockDim.x + threadIdx.x;
  for (int i = (nchunks << 5) + gid; i < n4; i += stride) {
    float4 v = x4[i];
    atomicAdd(&h[hobs_bin(v.x, mn, rng, ok)], 1u);
    atomicAdd(&h[hobs_bin(v.y, mn, rng, ok)], 1u);
    atomicAdd(&h[hobs_bin(v.z, mn, rng, ok)], 1u);
    atomicAdd(&h[hobs_bin(v.w, mn, rng, ok)], 1u);
  }
  for (int i = (n4 << 2) + gid; i < n; i += stride) {
    atomicAdd(&h[hobs_bin(x[i], mn, rng, ok)], 1u);
  }

  asm volatile("s_wait_asynccnt 0x0" ::: "memory");      // defensive drain
  __syncthreads();
  for (int i = threadIdx.x; i < BINS; i += TPB) {
    unsigned cnt = h[i];
    if (cnt) atomicAdd(&ws[i], cnt);             // sparse flush, global_atomic_add_u32
  }
}

__global__ void hobs_combine_kernel(const float* __restrict__ prev_hist,
                                    const float* __restrict__ prev_min,
                                    const float* __restrict__ prev_max,
                                    const unsigned* __restrict__ ws,
                                    float* __restrict__ out) {
  __shared__ float acc[BINS];
  for (int i = threadIdx.x; i < BINS; i += TPB) acc[i] = 0.0f;
  const float new_min = key2f(ws[BINS]);
  const float new_max = key2f(ws[BINS + 1]);
  const float pmin = prev_min[0];
  const float pmax = prev_max[0];
  const float cmin = fminf(new_min, pmin);
  const float cmax = fmaxf(new_max, pmax);
  const float dst_w = (cmax - cmin) / (float)BINS;
  const float safe_dw = (dst_w != 0.0f) ? dst_w : 1.0f;
  __syncthreads();
  #pragma unroll
  for (int s = 0; s < 2; ++s) {
    const float smn = s ? new_min : pmin;
    const float smx = s ? new_max : pmax;
    const float src_w = (smx - smn) / (float)BINS;
    const bool deg = (src_w == 0.0f) || (dst_w == 0.0f);
    const float safe_sw = (src_w != 0.0f) ? src_w : 1.0f;
    for (int i = threadIdx.x; i < BINS; i += TPB) {
      const float cnt = s ? (float)ws[i] : prev_hist[i];
      const float begin = smn + src_w * (float)i;
      const float end = begin + src_w;
      const float db  = (dst_w != 0.0f) ? floorf((begin - cmin) / safe_dw) : 0.0f;
      const float db2 = (dst_w != 0.0f)
                          ? fminf(floorf((end - cmin) / safe_dw), (float)(BINS - 1))
                          : 0.0f;
      const float dst_bin_end = cmin + dst_w * (db + 1.0f);
      const float frac = rintf((dst_bin_end - begin) / safe_sw * cnt);  // RNE == jnp.round
      const float c1 = deg ? cnt : fminf(frac, cnt);
      const float c2 = cnt - c1;
      int ib = (int)db;   ib  = ib  < 0 ? 0 : (ib  > BINS - 1 ? BINS - 1 : ib);
      int ib2 = (int)db2; ib2 = ib2 < 0 ? 0 : (ib2 > BINS - 1 ? BINS - 1 : ib2);
      atomicAdd(&acc[ib], c1);     // ds_add_f32
      atomicAdd(&acc[ib2], c2);    // ds_add_f32
    }
  }
  __syncthreads();
  for (int i = threadIdx.x; i < BINS; i += TPB) out[i] = acc[i];
  if (threadIdx.x == 0) { out[BINS] = cmin; out[BINS + 1] = cmax; }
}

extern "C" void kernel_launch(void* const* d_in, const int* in_sizes, int n_in,
                              void* d_out, int out_size, void* d_ws, size_t ws_size,
                              hipStream_t stream) {
  const float* x         = (const float*)d_in[0];
  const float* prev_hist = (const float*)d_in[1];
  const float* prev_min  = (const float*)d_in[2];
  const float* prev_max  = (const float*)d_in[3];
  float* out = (float*)d_out;
  unsigned* ws = (unsigned*)d_ws;

  const int n  = in_sizes[0];
  const int n4 = n >> 2;
  int blocks = (n4 + TPB - 1) / TPB;
  if (blocks > MAXBLOCKS) blocks = MAXBLOCKS;
  if (blocks < 1) blocks = 1;

  hipLaunchKernelGGL(hobs_init_kernel, dim3(1), dim3(TPB), 0, stream, ws);
  hipLaunchKernelGGL(hobs_minmax_kernel, dim3(blocks), dim3(TPB), 0, stream,
                     (const float4*)x, n4, x, n, ws);
  hipLaunchKernelGGL(hobs_hist_kernel, dim3(blocks), dim3(TPB), 0, stream,
                     (const float4*)x, n4, x, n, ws);
  hipLaunchKernelGGL(hobs_combine_kernel, dim3(1), dim3(TPB), 0, stream,
                     prev_hist, prev_min, prev_max, ws, out);
}